// NystromAttention_84610855731564
// MI455X (gfx1250) — compile-verified
//
#include <hip/hip_runtime.h>

typedef __attribute__((ext_vector_type(2))) float v2f;
typedef __attribute__((ext_vector_type(8))) float v8f;

static __device__ __forceinline__ v8f wmma4(v2f a, v2f b, v8f c) {
  // V_WMMA_F32_16X16X4_F32: D = A(16x4) * B(4x16) + C(16x16), all f32
  return __builtin_amdgcn_wmma_f32_16x16x4_f32(false, a, false, b, (short)0, c,
                                               false, false);
}

// ---------------------------------------------------------------------------
// Async global->LDS staging (GLOBAL_LOAD_ASYNC_TO_LDS_*, ASYNCcnt-tracked).
// Builtin signatures (from hipcc diagnostics):
//   b32 : (int AS1*, int AS3*, imm offset, imm cpol)
//   b128: (vector_size(16) int AS1*, vector_size(16) int AS3*, imm, imm)
// ---------------------------------------------------------------------------
#if defined(__has_builtin)
#if __has_builtin(__builtin_amdgcn_global_load_async_to_lds_b32)
#define NY_ASYNC32 1
#endif
#if __has_builtin(__builtin_amdgcn_global_load_async_to_lds_b128)
#define NY_ASYNC128 1
#endif
#endif

typedef int ny_v4i __attribute__((vector_size(16)));
typedef __attribute__((address_space(1))) int ny_gi32;
typedef __attribute__((address_space(3))) int ny_li32;
typedef __attribute__((address_space(1))) ny_v4i ny_gv4i;
typedef __attribute__((address_space(3))) ny_v4i ny_lv4i;

static __device__ __forceinline__ void cp_f32(const float* g, float* l) {
#ifdef NY_ASYNC32
  __builtin_amdgcn_global_load_async_to_lds_b32(
      (ny_gi32*)(uintptr_t)g, (ny_li32*)(uintptr_t)l, 0, 0);
#else
  *l = *g;
#endif
}

static __device__ __forceinline__ void cp_f32x4(const float* g, float* l) {
#ifdef NY_ASYNC128
  __builtin_amdgcn_global_load_async_to_lds_b128(
      (ny_gv4i*)(uintptr_t)g, (ny_lv4i*)(uintptr_t)l, 0, 0);
#else
  *(float4*)l = *(const float4*)g;
#endif
}

static __device__ __forceinline__ void cp_wait() {
#if defined(NY_ASYNC32) || defined(NY_ASYNC128)
  asm volatile("s_wait_asynccnt 0x0" ::: "memory");
#endif
}

// ---------------------------------------------------------------------------
// QKV GEMM: seq(b,n,c) = transpose(x);  qkv = seq @ w_qkv (512 x 1536)
// Block tile 128x64, wave tile 32x32 (2x2 accumulators).
// ---------------------------------------------------------------------------
__global__ void __launch_bounds__(256)
nystrom_qkv_gemm(const float* __restrict__ x, const float* __restrict__ wqkv,
                 float* __restrict__ q, float* __restrict__ k,
                 float* __restrict__ v) {
  __shared__ __align__(16) float As[128][20];
  __shared__ __align__(16) float Bs[16][64];
  const int t = threadIdx.x;
  const int n0 = blockIdx.x * 64;   // column in [0,1536)
  const int r0 = blockIdx.y * 128;  // row in [0,32768)
  const int b = r0 >> 10;
  const int i0 = r0 & 1023;
  const int wave = t >> 5, lane = t & 31;
  const int mrow = (wave >> 1) * 32;
  const int ncolw = (wave & 1) * 32;
  const int l15 = lane & 15, lh = lane >> 4;
  const float* xb = x + (size_t)b * 512 * 1024;
  v8f acc[2][2] = {};
  for (int k0 = 0; k0 < 512; k0 += 16) {
    __syncthreads();
    {  // A: 128 rows x 16 k, gathered transposed from x (coalesced along m)
      const int m = t & 127, kslot = t >> 7;
#pragma unroll
      for (int rep = 0; rep < 8; ++rep) {
        const int kk = rep * 2 + kslot;
        cp_f32(&xb[(size_t)(k0 + kk) * 1024 + i0 + m], &As[m][kk]);
      }
    }
    {  // B: 16 x 64 tile of w_qkv
      const int n4 = (t & 15) * 4, kk = t >> 4;
      cp_f32x4(&wqkv[(size_t)(k0 + kk) * 1536 + n0 + n4], &Bs[kk][n4]);
    }
    cp_wait();
    __syncthreads();
#pragma unroll
    for (int ks = 0; ks < 4; ++ks) {
      const int kb = ks * 4 + lh * 2;
      v2f a0 = *(const v2f*)&As[mrow + l15][kb];
      v2f a1 = *(const v2f*)&As[mrow + 16 + l15][kb];
      v2f b0, b1;
      b0.x = Bs[kb][ncolw + l15];
      b0.y = Bs[kb + 1][ncolw + l15];
      b1.x = Bs[kb][ncolw + 16 + l15];
      b1.y = Bs[kb + 1][ncolw + 16 + l15];
      acc[0][0] = wmma4(a0, b0, acc[0][0]);
      acc[0][1] = wmma4(a0, b1, acc[0][1]);
      acc[1][0] = wmma4(a1, b0, acc[1][0]);
      acc[1][1] = wmma4(a1, b1, acc[1][1]);
    }
  }
  const int mb = lh * 8;
#pragma unroll
  for (int ct = 0; ct < 2; ++ct) {
    const int jb = n0 + ncolw + ct * 16;  // wave-uniform
    const int sel = __builtin_amdgcn_readfirstlane(jb >> 9);
    const int h = __builtin_amdgcn_readfirstlane((jb & 511) >> 6);
    const int d = (jb & 63) + l15;
    float* dst = (sel == 0) ? q : (sel == 1 ? k : v);
    const float scale = (sel == 0) ? 0.125f : 1.0f;  // d^-0.5 on q
#pragma unroll
    for (int rt = 0; rt < 2; ++rt) {
#pragma unroll
      for (int r = 0; r < 8; ++r) {
        const int i = i0 + mrow + rt * 16 + mb + r;
        dst[(((size_t)b * 8 + h) * 1024 + i) * 64 + d] = acc[rt][ct][r] * scale;
      }
    }
  }
}

// ---------------------------------------------------------------------------
// Landmark means: ql/kl (bh,256,64) = mean over groups of 4 rows of q/k
// ---------------------------------------------------------------------------
__global__ void __launch_bounds__(256)
nystrom_landmarks(const float* __restrict__ q, const float* __restrict__ k,
                  float* __restrict__ ql, float* __restrict__ kl) {
  const size_t e = (size_t)blockIdx.x * 256 + threadIdx.x;  // 4194304
  const int d = e & 63;
  const size_t rest = e >> 6;
  const int mm = rest & 255;
  const size_t bh = rest >> 8;
  const size_t base = (bh * 1024 + (size_t)mm * 4) * 64 + d;
  ql[e] = 0.25f * (q[base] + q[base + 64] + q[base + 128] + q[base + 192]);
  kl[e] = 0.25f * (k[base] + k[base + 64] + k[base + 128] + k[base + 192]);
}

// ---------------------------------------------------------------------------
// Batched C = A @ B^T  (A: M x 64 rows, B: N x 64 rows, per bh), K = 64.
// B^T fragments are contiguous row pairs, read straight from global.
// ---------------------------------------------------------------------------
__global__ void __launch_bounds__(256)
nystrom_bgemm_nt(const float* __restrict__ A, const float* __restrict__ Bm,
                 float* __restrict__ C, int M, int N, size_t sA, size_t sB,
                 size_t sC) {
  __shared__ __align__(16) float As[128][20];
  const int t = threadIdx.x;
  const int bh = blockIdx.z;
  const float* Ab = A + (size_t)bh * sA;
  const float* Bb = Bm + (size_t)bh * sB;
  float* Cb = C + (size_t)bh * sC;
  const int n0 = blockIdx.x * 64, m0 = blockIdx.y * 128;
  const int wave = t >> 5, lane = t & 31;
  const int mrow = (wave >> 1) * 32;
  const int ncolw = (wave & 1) * 32;
  const int l15 = lane & 15, lh = lane >> 4;
  v8f acc[2][2] = {};
  for (int k0 = 0; k0 < 64; k0 += 16) {
    __syncthreads();
    {
      const int m = t >> 1, ks8 = (t & 1) * 8;
      const float* src = Ab + (size_t)(m0 + m) * 64 + k0 + ks8;
      cp_f32x4(src, &As[m][ks8]);
      cp_f32x4(src + 4, &As[m][ks8 + 4]);
    }
    cp_wait();
    __syncthreads();
    const float* Br0 = Bb + (size_t)(n0 + ncolw + l15) * 64 + k0;
    const float* Br1 = Bb + (size_t)(n0 + ncolw + 16 + l15) * 64 + k0;
#pragma unroll
    for (int ks = 0; ks < 4; ++ks) {
      const int kb = ks * 4 + lh * 2;
      v2f a0 = *(const v2f*)&As[mrow + l15][kb];
      v2f a1 = *(const v2f*)&As[mrow + 16 + l15][kb];
      v2f b0 = *(const v2f*)(Br0 + kb);
      v2f b1 = *(const v2f*)(Br1 + kb);
      acc[0][0] = wmma4(a0, b0, acc[0][0]);
      acc[0][1] = wmma4(a0, b1, acc[0][1]);
      acc[1][0] = wmma4(a1, b0, acc[1][0]);
      acc[1][1] = wmma4(a1, b1, acc[1][1]);
    }
  }
  const int mb = lh * 8;
#pragma unroll
  for (int rt = 0; rt < 2; ++rt) {
#pragma unroll
    for (int ct = 0; ct < 2; ++ct) {
#pragma unroll
      for (int r = 0; r < 8; ++r) {
        const int row = m0 + mrow + rt * 16 + mb + r;
        Cb[(size_t)row * N + n0 + ncolw + ct * 16 + l15] = acc[rt][ct][r];
      }
    }
  }
}

// ---------------------------------------------------------------------------
// Batched D = alpha*(A @ B) + aw*A [+ D]   (A: MxK, B: KxN, per bh)
// aw path used only for the square pinv updates (N == K).
// ---------------------------------------------------------------------------
__global__ void __launch_bounds__(256)
nystrom_bgemm_nn(const float* __restrict__ A, const float* __restrict__ Bm,
                 float* __restrict__ C, int M, int N, int K, size_t sA,
                 size_t sB, size_t sC, float alpha, float aw, int accum) {
  __shared__ __align__(16) float As[128][20];
  __shared__ __align__(16) float Bs[16][64];
  const int t = threadIdx.x;
  const int bh = blockIdx.z;
  const float* Ab = A + (size_t)bh * sA;
  const float* Bb = Bm + (size_t)bh * sB;
  float* Cb = C + (size_t)bh * sC;
  const int n0 = blockIdx.x * 64, m0 = blockIdx.y * 128;
  const int wave = t >> 5, lane = t & 31;
  const int mrow = (wave >> 1) * 32;
  const int ncolw = (wave & 1) * 32;
  const int l15 = lane & 15, lh = lane >> 4;
  v8f acc[2][2] = {};
  for (int k0 = 0; k0 < K; k0 += 16) {
    __syncthreads();
    {
      const int m = t >> 1, ks8 = (t & 1) * 8;
      const float* src = Ab + (size_t)(m0 + m) * K + k0 + ks8;
      cp_f32x4(src, &As[m][ks8]);
      cp_f32x4(src + 4, &As[m][ks8 + 4]);
    }
    {
      const int n4 = (t & 15) * 4, kk = t >> 4;
      cp_f32x4(&Bb[(size_t)(k0 + kk) * N + n0 + n4], &Bs[kk][n4]);
    }
    cp_wait();
    __syncthreads();
#pragma unroll
    for (int ks = 0; ks < 4; ++ks) {
      const int kb = ks * 4 + lh * 2;
      v2f a0 = *(const v2f*)&As[mrow + l15][kb];
      v2f a1 = *(const v2f*)&As[mrow + 16 + l15][kb];
      v2f b0, b1;
      b0.x = Bs[kb][ncolw + l15];
      b0.y = Bs[kb + 1][ncolw + l15];
      b1.x = Bs[kb][ncolw + 16 + l15];
      b1.y = Bs[kb + 1][ncolw + 16 + l15];
      acc[0][0] = wmma4(a0, b0, acc[0][0]);
      acc[0][1] = wmma4(a0, b1, acc[0][1]);
      acc[1][0] = wmma4(a1, b0, acc[1][0]);
      acc[1][1] = wmma4(a1, b1, acc[1][1]);
    }
  }
  const int mb = lh * 8;
#pragma unroll
  for (int rt = 0; rt < 2; ++rt) {
#pragma unroll
    for (int ct = 0; ct < 2; ++ct) {
#pragma unroll
      for (int r = 0; r < 8; ++r) {
        const int row = m0 + mrow + rt * 16 + mb + r;
        const int col = n0 + ncolw + ct * 16 + l15;
        float val = alpha * acc[rt][ct][r];
        if (aw != 0.0f) val += aw * Ab[(size_t)row * K + col];  // square only
        const size_t idx = (size_t)row * N + col;
        if (accum) val += Cb[idx];
        Cb[idx] = val;
      }
    }
  }
}

// ---------------------------------------------------------------------------
// Row softmax, in place.  One block per row.
// ---------------------------------------------------------------------------
__global__ void __launch_bounds__(256)
nystrom_softmax_rows(float* __restrict__ data, int len) {
  __shared__ float red[256];
  float* p = data + (size_t)blockIdx.x * len;
  const int t = threadIdx.x;
  float m = -3.0e38f;
  for (int j = t; j < len; j += 256) m = fmaxf(m, p[j]);
  red[t] = m;
  __syncthreads();
  for (int s = 128; s > 0; s >>= 1) {
    if (t < s) red[t] = fmaxf(red[t], red[t + s]);
    __syncthreads();
  }
  m = red[0];
  __syncthreads();
  float sum = 0.0f;
  for (int j = t; j < len; j += 256) {
    const float e = expf(p[j] - m);
    p[j] = e;
    sum += e;
  }
  red[t] = sum;
  __syncthreads();
  for (int s = 128; s > 0; s >>= 1) {
    if (t < s) red[t] += red[t + s];
    __syncthreads();
  }
  const float inv = 1.0f / red[0];
  for (int j = t; j < len; j += 256) p[j] *= inv;
}

// ---------------------------------------------------------------------------
// pinv init helpers
// ---------------------------------------------------------------------------
__global__ void nystrom_zero_scal(unsigned int* s) {
  if (threadIdx.x < 2) s[threadIdx.x] = 0u;
}

__global__ void __launch_bounds__(256)
nystrom_absmax(const float* __restrict__ attn2,
               unsigned int* __restrict__ scal) {
  __shared__ float red[256];
  const float* p = attn2 + (size_t)blockIdx.x * 65536;
  const int t = threadIdx.x;
  float rs = 0.0f, cs = 0.0f;
  for (int j = 0; j < 256; ++j) {
    rs += fabsf(p[(size_t)t * 256 + j]);  // sum over last axis
    cs += fabsf(p[(size_t)j * 256 + t]);  // sum over axis -2
  }
  red[t] = rs;
  __syncthreads();
  for (int s = 128; s > 0; s >>= 1) {
    if (t < s) red[t] = fmaxf(red[t], red[t + s]);
    __syncthreads();
  }
  if (t == 0) atomicMax(scal + 0, __float_as_uint(red[0]));
  __syncthreads();
  red[t] = cs;
  __syncthreads();
  for (int s = 128; s > 0; s >>= 1) {
    if (t < s) red[t] = fmaxf(red[t], red[t + s]);
    __syncthreads();
  }
  if (t == 0) atomicMax(scal + 1, __float_as_uint(red[0]));
}

__global__ void __launch_bounds__(256)
nystrom_pinv_init(const float* __restrict__ attn2,
                  const unsigned int* __restrict__ scal,
                  float* __restrict__ z) {
  const size_t e = (size_t)blockIdx.x * 256 + threadIdx.x;  // 16777216
  const float s = 1.0f / (__uint_as_float(scal[0]) * __uint_as_float(scal[1]));
  const size_t bh = e >> 16;
  const int ij = (int)(e & 65535);
  const int i = ij >> 8, j = ij & 255;
  z[e] = attn2[(bh << 16) + ((size_t)j << 8) + i] * s;  // transpose * scale
}

// ---------------------------------------------------------------------------
// Depthwise residual conv over sequence dim (K=33, same padding): outh=conv(v)
// ---------------------------------------------------------------------------
__global__ void __launch_bounds__(256)
nystrom_conv_res(const float* __restrict__ v, const float* __restrict__ kern,
                 float* __restrict__ outh) {
  const size_t e = (size_t)blockIdx.x * 256 + threadIdx.x;  // 16777216
  const int d = e & 63;
  const size_t rest = e >> 6;
  const int i = rest & 1023;
  const size_t bh = rest >> 10;
  const int h = bh & 7;
  const float* kr = kern + h * 33;
  const float* vb = v + (bh << 16);
  float s = 0.0f;
#pragma unroll
  for (int tp = 0; tp < 33; ++tp) {
    const int ii = i + tp - 16;
    if (ii >= 0 && ii < 1024) s += vb[(size_t)ii * 64 + d] * kr[tp];
  }
  outh[e] = s;
}

// ---------------------------------------------------------------------------
// Output GEMM: gather heads (b,h,n,d)->(b*n, 512), @ w_out + b_out,
// scatter to NCHW (b, c, H, W)
// ---------------------------------------------------------------------------
__global__ void __launch_bounds__(256)
nystrom_out_gemm(const float* __restrict__ outh, const float* __restrict__ wout,
                 const float* __restrict__ bout, float* __restrict__ out) {
  __shared__ __align__(16) float As[128][20];
  __shared__ __align__(16) float Bs[16][64];
  const int t = threadIdx.x;
  const int n0 = blockIdx.x * 64;   // c in [0,512)
  const int r0 = blockIdx.y * 128;  // row in [0,32768)
  const int b = r0 >> 10;
  const int i0 = r0 & 1023;
  const int wave = t >> 5, lane = t & 31;
  const int mrow = (wave >> 1) * 32;
  const int ncolw = (wave & 1) * 32;
  const int l15 = lane & 15, lh = lane >> 4;
  v8f acc[2][2] = {};
  for (int k0 = 0; k0 < 512; k0 += 16) {
    __syncthreads();
    {  // A: gather heads back to (b*n, 512) rows
      const int m = t >> 1, ks8 = (t & 1) * 8;
      const int kA = k0 + ks8;
      const int h = kA >> 6, d = kA & 63;
      const float* src =
          outh + (((size_t)b * 8 + h) * 1024 + i0 + m) * 64 + d;
      cp_f32x4(src, &As[m][ks8]);
      cp_f32x4(src + 4, &As[m][ks8 + 4]);
    }
    {
      const int n4 = (t & 15) * 4, kk = t >> 4;
      cp_f32x4(&wout[(size_t)(k0 + kk) * 512 + n0 + n4], &Bs[kk][n4]);
    }
    cp_wait();
    __syncthreads();
#pragma unroll
    for (int ks = 0; ks < 4; ++ks) {
      const int kb = ks * 4 + lh * 2;
      v2f a0 = *(const v2f*)&As[mrow + l15][kb];
      v2f a1 = *(const v2f*)&As[mrow + 16 + l15][kb];
      v2f b0, b1;
      b0.x = Bs[kb][ncolw + l15];
      b0.y = Bs[kb + 1][ncolw + l15];
      b1.x = Bs[kb][ncolw + 16 + l15];
      b1.y = Bs[kb + 1][ncolw + 16 + l15];
      acc[0][0] = wmma4(a0, b0, acc[0][0]);
      acc[0][1] = wmma4(a0, b1, acc[0][1]);
      acc[1][0] = wmma4(a1, b0, acc[1][0]);
      acc[1][1] = wmma4(a1, b1, acc[1][1]);
    }
  }
  const int mb = lh * 8;
#pragma unroll
  for (int rt = 0; rt < 2; ++rt) {
#pragma unroll
    for (int ct = 0; ct < 2; ++ct) {
#pragma unroll
      for (int r = 0; r < 8; ++r) {
        const int i = i0 + mrow + rt * 16 + mb + r;
        const int c = n0 + ncolw + ct * 16 + l15;
        out[((size_t)b * 512 + c) * 1024 + i] = acc[rt][ct][r] + bout[c];
      }
    }
  }
}

// ---------------------------------------------------------------------------
extern "C" void kernel_launch(void* const* d_in, const int* in_sizes, int n_in,
                              void* d_out, int out_size, void* d_ws,
                              size_t ws_size, hipStream_t stream) {
  (void)in_sizes; (void)n_in; (void)out_size; (void)ws_size;
  const float* x = (const float*)d_in[0];
  const float* wqkv = (const float*)d_in[1];
  const float* wout = (const float*)d_in[2];
  const float* bout = (const float*)d_in[3];
  const float* rker = (const float*)d_in[4];
  float* out = (float*)d_out;
  float* ws = (float*)d_ws;

  // workspace arena (floats)
  size_t o = 0;
  float* q     = ws + o; o += 16777216;  // (bh,1024,64)
  float* kmat  = ws + o; o += 16777216;
  float* v     = ws + o; o += 16777216;
  float* ql    = ws + o; o += 4194304;   // (bh,256,64)
  float* kl    = ws + o; o += 4194304;
  float* attn1 = ws + o; o += 67108864;  // (bh,1024,256)
  float* attn2 = ws + o; o += 16777216;  // (bh,256,256)
  float* attn3 = ws + o; o += 67108864;  // (bh,256,1024)
  float* kv    = ws + o; o += 4194304;   // (bh,256,64)
  float* kv2   = ws + o; o += 4194304;
  float* outh  = ws + o; o += 16777216;  // (bh,1024,64)
  unsigned int* scal = (unsigned int*)(ws + o); o += 16;
  // pinv scratch aliases regions already consumed by the time it runs
  float* z  = q;                   // q dead after sims
  float* z2 = kmat;                // k dead after sims
  float* xz = attn3;               // attn3 dead after kv
  float* wb = attn3 + 16777216;
  float* w2 = attn3 + 33554432;

  // 1) QKV projection (fused NCHW->seq transpose, head split, q scaling)
  nystrom_qkv_gemm<<<dim3(24, 256), 256, 0, stream>>>(x, wqkv, q, kmat, v);
  // 2) landmark means
  nystrom_landmarks<<<16384, 256, 0, stream>>>(q, kmat, ql, kl);
  // 3) similarity GEMMs (A @ B^T, K=64)
  nystrom_bgemm_nt<<<dim3(4, 8, 256), 256, 0, stream>>>(
      q, kl, attn1, 1024, 256, 65536, 16384, 262144);
  nystrom_bgemm_nt<<<dim3(4, 2, 256), 256, 0, stream>>>(
      ql, kl, attn2, 256, 256, 16384, 16384, 65536);
  nystrom_bgemm_nt<<<dim3(16, 2, 256), 256, 0, stream>>>(
      ql, kmat, attn3, 256, 1024, 16384, 65536, 262144);
  // 4) softmaxes
  nystrom_softmax_rows<<<262144, 256, 0, stream>>>(attn1, 256);
  nystrom_softmax_rows<<<65536, 256, 0, stream>>>(attn2, 256);
  nystrom_softmax_rows<<<65536, 256, 0, stream>>>(attn3, 1024);
  // 5) kv = attn3 @ v   (before pinv so attn3's space can be reused)
  nystrom_bgemm_nn<<<dim3(1, 2, 256), 256, 0, stream>>>(
      attn3, v, kv, 256, 64, 1024, 262144, 65536, 16384, 1.0f, 0.0f, 0);
  // 6) Moore-Penrose pinv of attn2 (6 Newton-Schulz iterations)
  nystrom_zero_scal<<<1, 32, 0, stream>>>(scal);
  nystrom_absmax<<<256, 256, 0, stream>>>(attn2, scal);
  nystrom_pinv_init<<<65536, 256, 0, stream>>>(attn2, scal, z);
  for (int it = 0; it < 6; ++it) {
    // xz = attn2 @ z
    nystrom_bgemm_nn<<<dim3(4, 2, 256), 256, 0, stream>>>(
        attn2, z, xz, 256, 256, 256, 65536, 65536, 65536, 1.0f, 0.0f, 0);
    // wb = xz @ (7I - xz) = 7*xz - xz@xz
    nystrom_bgemm_nn<<<dim3(4, 2, 256), 256, 0, stream>>>(
        xz, xz, wb, 256, 256, 256, 65536, 65536, 65536, -1.0f, 7.0f, 0);
    // w2 = xz @ (15I - wb) = 15*xz - xz@wb
    nystrom_bgemm_nn<<<dim3(4, 2, 256), 256, 0, stream>>>(
        xz, wb, w2, 256, 256, 256, 65536, 65536, 65536, -1.0f, 15.0f, 0);
    // z2 = 0.25 * z @ (13I - w2) = 3.25*z - 0.25*z@w2
    nystrom_bgemm_nn<<<dim3(4, 2, 256), 256, 0, stream>>>(
        z, w2, z2, 256, 256, 256, 65536, 65536, 65536, -0.25f, 3.25f, 0);
    float* tmp = z; z = z2; z2 = tmp;
  }
  // 7) kv2 = pinv(attn2) @ kv
  nystrom_bgemm_nn<<<dim3(1, 2, 256), 256, 0, stream>>>(
      z, kv, kv2, 256, 64, 256, 65536, 16384, 16384, 1.0f, 0.0f, 0);
  // 8) outh = depthwise_conv(v); then outh += attn1 @ kv2
  nystrom_conv_res<<<65536, 256, 0, stream>>>(v, rker, outh);
  nystrom_bgemm_nn<<<dim3(1, 8, 256), 256, 0, stream>>>(
      attn1, kv2, outh, 1024, 64, 256, 262144, 16384, 65536, 1.0f, 0.0f, 1);
  // 9) output projection + bias, scatter to NCHW
  nystrom_out_gemm<<<dim3(8, 256), 256, 0, stream>>>(outh, wout, bout, out);
}